// GPT_4209067950358
// MI455X (gfx1250) — compile-verified
//
#include <hip/hip_runtime.h>
#include <math.h>

#define Bq 2
#define Tq 1024
#define Cq 768
#define Vq 32000
#define Lq 4
#define NBq 4
#define KCONV 15
#define Pq 16
#define MRq (Bq*Tq)       // 2048 rows
#define NBC (NBq*Cq)      // 3072
#define HALF_C (Cq/2)     // 384

#define SCALEq 1.8137993642342178f   // pi/sqrt(3)
#define EPSq   1.1920929e-07f

typedef float v2f __attribute__((ext_vector_type(2)));
typedef float v8f __attribute__((ext_vector_type(8)));

__device__ __forceinline__ float lelu_f(float x) {
  return x / (1.f + expf(-SCALEq * x));
}
__device__ __forceinline__ float softplus_f(float x) {
  return (x > 20.f) ? x : log1pf(expf(x));
}

// ---------------------------------------------------------------------------
// Generic batched GEMM: Out[z] = epi(A[z] @ B[z] (+ Res[z]))
// Per-operand batch offset = (z/zDiv)*S1 + (z%zDiv)*S2 (handles (b,n) batching).
// bTrans=1 means B[k][n] = Bm[n*ldb + k]  (used for Q @ K^T).
// epi: 0 = none, 1 = lelu, 2 = add residual.
//
// Block = 128 threads = 4 waves (wave32); block tile 64x64, wave tile 32x32
// (2x2 of 16x16 WMMA frags), K-step 16 (4 x V_WMMA_F32_16X16X4_F32 per frag).
// REQUIRES: M,N multiples of 64 and K multiple of 16 (true for every launch).
// Double-buffered LDS; tiles staged through registers with b128 global loads
// so the next tile's vmem latency overlaps the current tile's WMMAs.
// ---------------------------------------------------------------------------
__global__ __launch_bounds__(128) void wmma_gemm_f32(
    const float* __restrict__ A,  long long aS1, long long aS2, int lda,
    const float* __restrict__ Bm, long long bS1, long long bS2, int ldb, int bTrans,
    float* __restrict__ Out,      long long oS1, long long oS2, int ldo,
    const float* __restrict__ Res,long long rS1, long long rS2, int ldr,
    int M, int N, int Kd, int zDiv, int epi)
{
  __shared__ float As[2][64][17];
  __shared__ float Bs[2][16][65];

  int z  = blockIdx.z;
  long long zq = z / zDiv, zr = z % zDiv;
  const float* Ab = A  + zq*aS1 + zr*aS2;
  const float* Bb = Bm + zq*bS1 + zr*bS2;
  float*       Ob = Out + zq*oS1 + zr*oS2;
  const float* Rb = (epi == 2) ? (Res + zq*rS1 + zr*rS2) : nullptr;

  int n0 = blockIdx.x * 64;
  int m0 = blockIdx.y * 64;
  int tid  = threadIdx.x;
  int wid  = tid >> 5;
  int lane = tid & 31;
  int r    = lane & 15;
  int hi   = lane >> 4;
  int wm = (wid >> 1) * 32;
  int wn = (wid & 1) * 32;

  // staging coordinates (all loads are 16B-aligned: every lda/ldb here is a
  // multiple of 4 floats and k/n offsets are multiples of 8 floats)
  int am  = tid >> 1, akp  = (tid & 1) * 8;   // A: row am, 8 consecutive k
  int bk  = tid >> 3, bnp  = (tid & 7) * 8;   // B (K-major): row bk, 8 consecutive n
  int btn = tid >> 1, btkp = (tid & 1) * 8;   // B (N-major): row btn, 8 consecutive k

  const long long aRow = (long long)(m0 + am) * lda + akp;
  float4 ra0{}, ra1{}, rb0{}, rb1{};

  auto loadTile = [&](int k0) {
    const float* pa = Ab + aRow + k0;
    ra0 = *reinterpret_cast<const float4*>(pa);
    ra1 = *reinterpret_cast<const float4*>(pa + 4);
    if (!bTrans) {
      const float* pb = Bb + (long long)(k0 + bk) * ldb + n0 + bnp;
      rb0 = *reinterpret_cast<const float4*>(pb);
      rb1 = *reinterpret_cast<const float4*>(pb + 4);
    } else {
      const float* pb = Bb + (long long)(n0 + btn) * ldb + k0 + btkp;
      rb0 = *reinterpret_cast<const float4*>(pb);
      rb1 = *reinterpret_cast<const float4*>(pb + 4);
    }
  };
  auto storeTile = [&](int buf) {
    float* ap = &As[buf][am][akp];
    ap[0] = ra0.x; ap[1] = ra0.y; ap[2] = ra0.z; ap[3] = ra0.w;
    ap[4] = ra1.x; ap[5] = ra1.y; ap[6] = ra1.z; ap[7] = ra1.w;
    if (!bTrans) {
      float* bp = &Bs[buf][bk][bnp];
      bp[0] = rb0.x; bp[1] = rb0.y; bp[2] = rb0.z; bp[3] = rb0.w;
      bp[4] = rb1.x; bp[5] = rb1.y; bp[6] = rb1.z; bp[7] = rb1.w;
    } else {
      Bs[buf][btkp + 0][btn] = rb0.x; Bs[buf][btkp + 1][btn] = rb0.y;
      Bs[buf][btkp + 2][btn] = rb0.z; Bs[buf][btkp + 3][btn] = rb0.w;
      Bs[buf][btkp + 4][btn] = rb1.x; Bs[buf][btkp + 5][btn] = rb1.y;
      Bs[buf][btkp + 6][btn] = rb1.z; Bs[buf][btkp + 7][btn] = rb1.w;
    }
  };

  v8f acc[2][2] = {};
  const int nsteps = Kd >> 4;

  loadTile(0);
  storeTile(0);
  __syncthreads();
  int buf = 0;

  for (int step = 0; step < nsteps; ++step) {
    if (step + 1 < nsteps) loadTile((step + 1) << 4);  // overlaps WMMA below

#if __has_builtin(__builtin_amdgcn_wmma_f32_16x16x4_f32)
    #pragma unroll
    for (int kk = 0; kk < 4; ++kk) {
      int kb = kk * 4 + 2 * hi;   // A frag: lanes 0-15 hold K={0,1}, 16-31 K={2,3}
      v2f af[2], bf[2];
      #pragma unroll
      for (int mi = 0; mi < 2; ++mi) {
        af[mi].x = As[buf][wm + mi * 16 + r][kb];
        af[mi].y = As[buf][wm + mi * 16 + r][kb + 1];
      }
      #pragma unroll
      for (int ni = 0; ni < 2; ++ni) {
        bf[ni].x = Bs[buf][kb][wn + ni * 16 + r];
        bf[ni].y = Bs[buf][kb + 1][wn + ni * 16 + r];
      }
      #pragma unroll
      for (int mi = 0; mi < 2; ++mi)
        #pragma unroll
        for (int ni = 0; ni < 2; ++ni)
          acc[mi][ni] = __builtin_amdgcn_wmma_f32_16x16x4_f32(
              false, af[mi], false, bf[ni], (short)0, acc[mi][ni], false, false);
    }
#else
    // VALU fallback (also what the host compilation pass parses)
    #pragma unroll
    for (int mi = 0; mi < 2; ++mi)
      #pragma unroll
      for (int ni = 0; ni < 2; ++ni) {
        int mbase = wm + mi * 16 + hi * 8;
        int n = wn + ni * 16 + r;
        for (int kk = 0; kk < 16; ++kk) {
          float bv = Bs[buf][kk][n];
          #pragma unroll
          for (int i = 0; i < 8; ++i)
            acc[mi][ni][i] += As[buf][mbase + i][kk] * bv;
        }
      }
#endif

    if (step + 1 < nsteps) {
      storeTile(buf ^ 1);
      __syncthreads();
      buf ^= 1;
    }
  }

  // store: C/D layout — VGPR i, lanes 0-15: M = i; lanes 16-31: M = 8+i; N = lane%16
  #pragma unroll
  for (int mi = 0; mi < 2; ++mi)
    #pragma unroll
    for (int ni = 0; ni < 2; ++ni)
      #pragma unroll
      for (int i = 0; i < 8; ++i) {
        int gm = m0 + wm + mi * 16 + hi * 8 + i;
        int gn = n0 + wn + ni * 16 + r;
        float v = acc[mi][ni][i];
        if (epi == 1)      v = lelu_f(v);
        else if (epi == 2) v += Rb[(long long)gm * ldr + gn];
        Ob[(long long)gm * ldo + gn] = v;
      }
}

// ---------------------------------------------------------------------------
// Small kernels
// ---------------------------------------------------------------------------
__global__ void embed_kernel(const int* __restrict__ idx, const float* __restrict__ wte,
                             float* __restrict__ x) {
  int row = blockIdx.x;
  int c = blockIdx.y * 256 + threadIdx.x;
  if (c < Cq) x[(long long)row * Cq + c] = wte[(long long)idx[row] * Cq + c];
}

__global__ void rope_tables_kernel(float* __restrict__ cosT, float* __restrict__ sinT) {
  int i = blockIdx.x * 256 + threadIdx.x;
  if (i >= Tq * HALF_C) return;
  int t = i / HALF_C, f = i % HALF_C;
  float inv = expf(-((2.f * f) / (float)Cq) * logf(10000.f));
  float fr = (float)t * inv;
  cosT[i] = cosf(fr);
  sinT[i] = sinf(fr);
}

__global__ __launch_bounds__(256) void rmsnorm_kernel(const float* __restrict__ in,
                                                      float* __restrict__ out, int ncol) {
  long long row = blockIdx.x;
  const float* p = in + row * ncol;
  float* o = out + row * ncol;
  float ss = 0.f;
  for (int c = threadIdx.x; c < ncol; c += 256) { float v = p[c]; ss += v * v; }
  __shared__ float red[256];
  red[threadIdx.x] = ss; __syncthreads();
  for (int s = 128; s > 0; s >>= 1) {
    if ((int)threadIdx.x < s) red[threadIdx.x] += red[threadIdx.x + s];
    __syncthreads();
  }
  float g = rsqrtf(red[0] / (float)ncol + EPSq);
  for (int c = threadIdx.x; c < ncol; c += 256) o[c] = p[c] * g;
}

__global__ void dwconv_kernel(const float* __restrict__ xn, const float* __restrict__ w,
                              float* __restrict__ xs) {
  int row = blockIdx.x; int b = row / Tq; int t = row % Tq;
  int c = blockIdx.y * 256 + threadIdx.x;
  if (c >= Cq) return;
  float acc = 0.f;
  #pragma unroll
  for (int j = 0; j < KCONV; ++j) {
    int tt = t - (KCONV - 1) + j;
    if (tt >= 0) acc += xn[((long long)(b * Tq + tt)) * Cq + c] * w[c * KCONV + j];
  }
  xs[(long long)row * Cq + c] = acc;
}

__global__ __launch_bounds__(128) void nav2_kernel(const float* __restrict__ h1,
                                                   const float* __restrict__ w2,
                                                   float* __restrict__ coords) {
  int row = blockIdx.x;
  float s0 = 0.f, s1 = 0.f;
  for (int c = threadIdx.x; c < HALF_C; c += 128) {
    float h = h1[(long long)row * HALF_C + c];
    s0 += h * w2[c * 2 + 0];
    s1 += h * w2[c * 2 + 1];
  }
  __shared__ float r0[128], r1[128];
  r0[threadIdx.x] = s0; r1[threadIdx.x] = s1; __syncthreads();
  for (int s = 64; s > 0; s >>= 1) {
    if ((int)threadIdx.x < s) { r0[threadIdx.x] += r0[threadIdx.x + s]; r1[threadIdx.x] += r1[threadIdx.x + s]; }
    __syncthreads();
  }
  if (threadIdx.x == 0) {
    coords[row * 2 + 0] = tanhf(r0[0]);
    coords[row * 2 + 1] = tanhf(r1[0]);
  }
}

__global__ void palette_kernel(const float* __restrict__ coords, const float* __restrict__ pal,
                               float* __restrict__ ret) {
  int row = blockIdx.x;
  int c = blockIdx.y * 256 + threadIdx.x;
  if (c >= Cq) return;
  float cx = coords[row * 2 + 0], cy = coords[row * 2 + 1];
  float ix = fminf(fmaxf((cx + 1.f) * 0.5f * (Pq - 1), 0.f), (float)(Pq - 1));
  float iy = fminf(fmaxf((cy + 1.f) * 0.5f * (Pq - 1), 0.f), (float)(Pq - 1));
  float x0f = floorf(ix), y0f = floorf(iy);
  float wx = ix - x0f, wy = iy - y0f;
  int x0 = (int)x0f, y0 = (int)y0f;
  int x1 = min(x0 + 1, Pq - 1), y1 = min(y0 + 1, Pq - 1);
  const float* pc = pal + (long long)c * Pq * Pq;
  float v = pc[y0 * Pq + x0] * (1.f - wx) * (1.f - wy)
          + pc[y0 * Pq + x1] * wx * (1.f - wy)
          + pc[y1 * Pq + x0] * (1.f - wx) * wy
          + pc[y1 * Pq + x1] * wx * wy;
  ret[(long long)row * Cq + c] = v;
}

// per-(b,n,t): rms-norm the 768-slice of Q, then RoPE (in place)
__global__ __launch_bounds__(256) void qnorm_rope_kernel(float* __restrict__ q,
                                                         const float* __restrict__ cosT,
                                                         const float* __restrict__ sinT) {
  int bid = blockIdx.x;
  int t = bid % Tq; int n = (bid / Tq) % NBq; int b = bid / (Tq * NBq);
  float* p = q + ((long long)(b * Tq + t)) * NBC + (long long)n * Cq;
  float ss = 0.f;
  for (int c = threadIdx.x; c < Cq; c += 256) { float v = p[c]; ss += v * v; }
  __shared__ float red[256];
  red[threadIdx.x] = ss; __syncthreads();
  for (int s = 128; s > 0; s >>= 1) {
    if ((int)threadIdx.x < s) red[threadIdx.x] += red[threadIdx.x + s];
    __syncthreads();
  }
  float g = rsqrtf(red[0] / (float)Cq + EPSq);
  for (int i = threadIdx.x; i < HALF_C; i += 256) {
    float x1 = p[2 * i] * g, x2 = p[2 * i + 1] * g;
    float cs = cosT[t * HALF_C + i], sn = sinT[t * HALF_C + i];
    p[2 * i]     = x1 * cs - x2 * sn;
    p[2 * i + 1] = x1 * sn + x2 * cs;
  }
}

__global__ void krope_kernel(float* __restrict__ k, const float* __restrict__ cosT,
                             const float* __restrict__ sinT) {
  int row = blockIdx.x; int t = row % Tq;
  float* p = k + (long long)row * Cq;
  for (int i = threadIdx.x; i < HALF_C; i += 256) {
    float x1 = p[2 * i], x2 = p[2 * i + 1];
    float cs = cosT[t * HALF_C + i], sn = sinT[t * HALF_C + i];
    p[2 * i]     = x1 * cs - x2 * sn;
    p[2 * i + 1] = x1 * sn + x2 * cs;
  }
}

// Converts raw scores att[b,n,t,s] (unscaled) in-place into comb = w * route,
// and produces routemax[b,n,t] and residual[b,t]. route == hard argmax over heads.
__global__ __launch_bounds__(256) void routing_kernel(float* __restrict__ att,
                                                      float* __restrict__ routemax,
                                                      float* __restrict__ resid) {
  int bid = blockIdx.x; int b = bid / Tq; int t = bid % Tq;
  const float invsq = 0.03608439182435161f;  // 1/sqrt(768)
  long long base = ((long long)b * NBq) * Tq * Tq + (long long)t * Tq;
  const long long hs = (long long)Tq * Tq;

  // pass 1: sum over s<=t of softplus(max_n att)
  float part = 0.f;
  for (int s = threadIdx.x; s <= t; s += 256) {
    float m = -3.0e38f;
    #pragma unroll
    for (int n = 0; n < NBq; ++n) m = fmaxf(m, att[base + n * hs + s] * invsq);
    part += softplus_f(m);
  }
  __shared__ float red[256];
  red[threadIdx.x] = part; __syncthreads();
  for (int s = 128; s > 0; s >>= 1) {
    if ((int)threadIdx.x < s) red[threadIdx.x] += red[threadIdx.x + s];
    __syncthreads();
  }
  float sumS = red[0];
  float wscale = fminf(1.f / (sumS + 1e-6f), 1.f);

  __shared__ int flags[NBq];
  if (threadIdx.x < NBq) flags[threadIdx.x] = 0;
  __syncthreads();

  int loc[NBq] = {0, 0, 0, 0};
  for (int s = threadIdx.x; s < Tq; s += 256) {
    if (s <= t) {
      float a_[NBq]; float m = -3.0e38f;
      #pragma unroll
      for (int n = 0; n < NBq; ++n) { a_[n] = att[base + n * hs + s] * invsq; m = fmaxf(m, a_[n]); }
      float w = softplus_f(m) * wscale;
      #pragma unroll
      for (int n = 0; n < NBq; ++n) {
        int rt = (a_[n] == m);
        att[base + n * hs + s] = rt ? w : 0.f;
        loc[n] |= rt;
      }
    } else {
      // masked: softplus(-inf)=0 for every head -> soft==max holds -> route=1, comb=0
      #pragma unroll
      for (int n = 0; n < NBq; ++n) { att[base + n * hs + s] = 0.f; loc[n] = 1; }
    }
  }
  #pragma unroll
  for (int n = 0; n < NBq; ++n) if (loc[n]) atomicOr(&flags[n], 1);
  __syncthreads();
  if (threadIdx.x == 0) resid[bid] = 1.f - sumS * wscale;
  if (threadIdx.x < NBq)
    routemax[((long long)b * NBq + threadIdx.x) * Tq + t] = (float)flags[threadIdx.x];
}

__global__ void combine_kernel(const float* __restrict__ ybn, const float* __restrict__ routemax,
                               const float* __restrict__ resid, const float* __restrict__ sb,
                               const float* __restrict__ sr, float* __restrict__ y) {
  int row = blockIdx.x; int b = row / Tq; int t = row % Tq;
  int c = blockIdx.y * 256 + threadIdx.x;
  if (c >= Cq) return;
  float acc = resid[row] * sr[c];
  #pragma unroll
  for (int n = 0; n < NBq; ++n) {
    long long hb = (long long)(b * NBq + n);
    acc += ybn[(hb * Tq + t) * Cq + c] + routemax[hb * Tq + t] * sb[n * Cq + c];
  }
  y[(long long)row * Cq + c] = acc;
}

__global__ void cumsum_kernel(const float* __restrict__ an, const float* __restrict__ u2,
                              float* __restrict__ xout) {
  int g = blockIdx.x * 256 + threadIdx.x;
  if (g >= Bq * Cq) return;
  int b = g / Cq, c = g % Cq;
  float acc = 0.f;
  for (int t = 0; t < Tq; ++t) {
    long long o = ((long long)(b * Tq + t)) * Cq + c;
    acc += an[o];
    xout[o] = u2[o] + acc / (float)(t + 1);
  }
}

// ---------------------------------------------------------------------------
extern "C" void kernel_launch(void* const* d_in, const int* in_sizes, int n_in,
                              void* d_out, int out_size, void* d_ws, size_t ws_size,
                              hipStream_t stream) {
  (void)in_sizes; (void)n_in; (void)out_size; (void)ws_size;
  const int*   idx       = (const int*)  d_in[0];
  const float* wte       = (const float*)d_in[1];
  const float* lm_head   = (const float*)d_in[2];
  const float* q_w       = (const float*)d_in[3];
  const float* k_w       = (const float*)d_in[4];
  const float* v_w       = (const float*)d_in[5];
  const float* o_w       = (const float*)d_in[6];
  const float* sink_res  = (const float*)d_in[7];
  const float* sink_basis= (const float*)d_in[8];
  const float* palette   = (const float*)d_in[9];
  const float* conv_w    = (const float*)d_in[10];
  const float* nav_w1    = (const float*)d_in[11];
  const float* nav_w2    = (const float*)d_in[12];
  const float* out_w1    = (const float*)d_in[13];
  const float* out_w2    = (const float*)d_in[14];
  const float* mlp1_fc   = (const float*)d_in[15];
  const float* mlp1_proj = (const float*)d_in[16];
  const float* mlp2_fc   = (const float*)d_in[17];
  const float* mlp2_proj = (const float*)d_in[18];
  float* out = (float*)d_out;
  float* ws  = (float*)d_ws;

  size_t off = 0;
  auto alloc = [&](size_t n) { float* p = ws + off; off += n; return p; };
  float* x0     = alloc((size_t)MRq * Cq);
  float* x1     = alloc((size_t)MRq * Cq);
  float* xn     = alloc((size_t)MRq * Cq);
  float* t0     = alloc((size_t)MRq * Cq);       // xs / ret / qn / a / un
  float* an     = alloc((size_t)MRq * Cq);
  float* h1     = alloc((size_t)MRq * HALF_C);
  float* coords = alloc((size_t)MRq * 2);
  float* t1     = alloc((size_t)MRq * 2 * Cq);   // h2 / y
  float* qman   = alloc((size_t)MRq * Cq);
  float* S1     = alloc((size_t)MRq * NBC);      // mlp hidden / Q / ybn
  float* S2     = alloc((size_t)MRq * NBC);      // V
  float* S3     = alloc((size_t)Bq * NBq * Tq * Tq);  // att -> comb
  float* kbuf   = alloc((size_t)MRq * Cq);       // K / u2
  float* ubuf   = alloc((size_t)MRq * Cq);
  float* cosT   = alloc((size_t)Tq * HALF_C);
  float* sinT   = alloc((size_t)Tq * HALF_C);
  float* routemax = alloc((size_t)Bq * NBq * Tq);
  float* residb   = alloc((size_t)Bq * Tq);

  auto gemm = [&](const float* A, long long aS1, long long aS2, int lda,
                  const float* Bm, long long bS1, long long bS2, int ldb, int bT,
                  float* O, long long oS1, long long oS2, int ldo,
                  const float* R, long long rS1, long long rS2, int ldr,
                  int M, int N, int Kd, int Z, int zDiv, int epi) {
    dim3 g(N / 64, M / 64, Z);
    wmma_gemm_f32<<<g, dim3(128), 0, stream>>>(A, aS1, aS2, lda, Bm, bS1, bS2, ldb, bT,
                                               O, oS1, oS2, ldo, R, rS1, rS2, ldr,
                                               M, N, Kd, zDiv, epi);
  };
  auto gemm1 = [&](const float* A, int lda, const float* Bm, int ldb, int bT,
                   float* O, int ldo, const float* R, int ldr,
                   int M, int N, int Kd, int epi) {
    gemm(A, 0, 0, lda, Bm, 0, 0, ldb, bT, O, 0, 0, ldo, R, 0, 0, ldr, M, N, Kd, 1, 1, epi);
  };

  rope_tables_kernel<<<(Tq * HALF_C + 255) / 256, 256, 0, stream>>>(cosT, sinT);
  embed_kernel<<<dim3(MRq, 3), 256, 0, stream>>>(idx, wte, x0);

  float* xcur = x0; float* xnext = x1;
  for (int l = 0; l < Lq; ++l) {
    const float* qw  = q_w       + (size_t)l * Cq * NBC;
    const float* kw  = k_w       + (size_t)l * Cq * Cq;
    const float* vw  = v_w       + (size_t)l * Cq * NBC;
    const float* ow  = o_w       + (size_t)l * Cq * Cq;
    const float* sr  = sink_res  + (size_t)l * Cq;
    const float* sb  = sink_basis+ (size_t)l * NBq * Cq;
    const float* pal = palette   + (size_t)l * Cq * Pq * Pq;
    const float* cw  = conv_w    + (size_t)l * Cq * KCONV;
    const float* nw1 = nav_w1    + (size_t)l * Cq * HALF_C;
    const float* nw2 = nav_w2    + (size_t)l * HALF_C * 2;
    const float* ow1 = out_w1    + (size_t)l * Cq * 2 * Cq;
    const float* ow2 = out_w2    + (size_t)l * 2 * Cq * Cq;
    const float* fc1 = mlp1_fc   + (size_t)l * Cq * 4 * Cq;
    const float* pj1 = mlp1_proj + (size_t)l * 4 * Cq * Cq;
    const float* fc2 = mlp2_fc   + (size_t)l * Cq * 4 * Cq;
    const float* pj2 = mlp2_proj + (size_t)l * 4 * Cq * Cq;

    // xn = norm(x)
    rmsnorm_kernel<<<MRq, 256, 0, stream>>>(xcur, xn, Cq);
    // ---- manifold -> q ----
    dwconv_kernel<<<dim3(MRq, 3), 256, 0, stream>>>(xn, cw, t0);                 // xs
    gemm1(t0, Cq, nw1, HALF_C, 0, h1, HALF_C, nullptr, 0, MRq, HALF_C, Cq, 1);   // lelu(xs@nw1)
    nav2_kernel<<<MRq, 128, 0, stream>>>(h1, nw2, coords);                       // tanh(.@nw2)
    palette_kernel<<<dim3(MRq, 3), 256, 0, stream>>>(coords, pal, t0);           // ret
    gemm1(t0, Cq, ow1, 2 * Cq, 0, t1, 2 * Cq, nullptr, 0, MRq, 2 * Cq, Cq, 1);   // lelu(ret@ow1)
    gemm1(t1, 2 * Cq, ow2, Cq, 0, qman, Cq, nullptr, 0, MRq, Cq, 2 * Cq, 0);     // q
    // ---- mlp1: a = q + lelu(norm(q)@fc)@proj ----
    rmsnorm_kernel<<<MRq, 256, 0, stream>>>(qman, t0, Cq);
    gemm1(t0, Cq, fc1, 4 * Cq, 0, S1, 4 * Cq, nullptr, 0, MRq, 4 * Cq, Cq, 1);
    gemm1(S1, 4 * Cq, pj1, Cq, 0, t0, Cq, qman, Cq, MRq, Cq, 4 * Cq, 2);         // a -> t0
    rmsnorm_kernel<<<MRq, 256, 0, stream>>>(t0, an, Cq);                         // an
    // ---- attention ----
    gemm1(an, Cq, qw, NBC, 0, S1, NBC, nullptr, 0, MRq, NBC, Cq, 0);             // Q
    qnorm_rope_kernel<<<Bq * NBq * Tq, 256, 0, stream>>>(S1, cosT, sinT);
    gemm1(an, Cq, vw, NBC, 0, S2, NBC, nullptr, 0, MRq, NBC, Cq, 0);             // V
    gemm1(xn, Cq, kw, Cq, 0, kbuf, Cq, nullptr, 0, MRq, Cq, Cq, 0);              // K
    krope_kernel<<<MRq, 256, 0, stream>>>(kbuf, cosT, sinT);
    // scores: att[b,n] = Qrope[b,n] @ Krope[b]^T   (batched over z=(b,n))
    gemm(S1, (long long)Tq * NBC, Cq, NBC,
         kbuf, (long long)Tq * Cq, 0, Cq, 1,
         S3, (long long)NBq * Tq * Tq, (long long)Tq * Tq, Tq,
         nullptr, 0, 0, 0,
         Tq, Tq, Cq, Bq * NBq, NBq, 0);
    routing_kernel<<<Bq * Tq, 256, 0, stream>>>(S3, routemax, residb);           // att -> comb
    // y_ctx[b,n] = comb[b,n] @ V[b,n]
    gemm(S3, (long long)NBq * Tq * Tq, (long long)Tq * Tq, Tq,
         S2, (long long)Tq * NBC, Cq, NBC, 0,
         S1, (long long)NBq * Tq * Cq, (long long)Tq * Cq, Cq,
         nullptr, 0, 0, 0,
         Tq, Cq, Tq, Bq * NBq, NBq, 0);
    combine_kernel<<<dim3(MRq, 3), 256, 0, stream>>>(S1, routemax, residb, sb, sr, t1); // y
    gemm1(t1, Cq, ow, Cq, 0, ubuf, Cq, xcur, Cq, MRq, Cq, Cq, 2);                // u = x + y@ow
    // ---- mlp2 ----
    rmsnorm_kernel<<<MRq, 256, 0, stream>>>(ubuf, t0, Cq);
    gemm1(t0, Cq, fc2, 4 * Cq, 0, S1, 4 * Cq, nullptr, 0, MRq, 4 * Cq, Cq, 1);
    gemm1(S1, 4 * Cq, pj2, Cq, 0, kbuf, Cq, ubuf, Cq, MRq, Cq, 4 * Cq, 2);       // u2 -> kbuf
    // ---- x = u2 + cumsum(an)/steps ----
    cumsum_kernel<<<(Bq * Cq + 255) / 256, 256, 0, stream>>>(an, kbuf, xnext);
    float* tmp = xcur; xcur = xnext; xnext = tmp;
  }

  // logits = norm(x) @ lm_head
  rmsnorm_kernel<<<MRq, 256, 0, stream>>>(xcur, xn, Cq);
  gemm1(xn, Cq, lm_head, Vq, 0, out, Vq, nullptr, 0, MRq, Vq, Cq, 0);
}